// CenterLoss_7928509628847
// MI455X (gfx1250) — compile-verified
//
#include <hip/hip_runtime.h>

// ---------------------------------------------------------------------------
// CenterLoss (MI455X / gfx1250, wave32):
//   loss = mean_i( 0.5 * || x[i] - centers[labels[i]] ||^2 )
// Pure bandwidth problem (~4.2 MB reads, ~1.6 MFLOP). f32 end to end.
// WMMA (V_WMMA_F32_16X16X4_F32 with all-ones B) is used as the in-wave
// cross-lane reduction tree.
// ---------------------------------------------------------------------------

typedef float v2f __attribute__((ext_vector_type(2)));
typedef float v8f __attribute__((ext_vector_type(8)));

#define FEAT  512
#define BATCH 1024

// Full 32-lane sum of per-lane partial p using one f32 WMMA.
// A(16x4): A[m,0]=p[m] (lanes 0-15, VGPR0), A[m,2]=p[m+16] (lanes 16-31, VGPR0),
//          K=1,3 columns zero.  B(4x16)=1  =>  D[m,n] = p[m] + p[m+16].
// D layout: VGPR v holds row M=v (lanes 0-15) / M=v+8 (lanes 16-31), so
// sum(c[0..7]) gives rows 0-7 on lanes<16 and rows 8-15 on lanes>=16;
// one xor-16 shuffle completes the total (broadcast to all lanes).
__device__ __forceinline__ float wave_reduce_sum_wmma(float p) {
  v2f a; a[0] = p;    a[1] = 0.0f;
  v2f b; b[0] = 1.0f; b[1] = 1.0f;   // all-ones regardless of B layout
  v8f c = {};
  c = __builtin_amdgcn_wmma_f32_16x16x4_f32(
      /*neg_a=*/false, a, /*neg_b=*/false, b,
      /*c_mod=*/(short)0, c, /*reuse_a=*/false, /*reuse_b=*/false);
  float s = ((c[0] + c[1]) + (c[2] + c[3])) + ((c[4] + c[5]) + (c[6] + c[7]));
  s += __shfl_xor(s, 16, 32);
  return s;
}

// One wave per sample row: 4x float4 loads from x-row and gathered center row,
// FMA diff^2 accumulate, WMMA cross-lane reduce, lane 0 stores raw row sumsq.
__global__ void __launch_bounds__(256)
center_loss_rows(const float* __restrict__ x,
                 const int*   __restrict__ labels,
                 const float* __restrict__ centers,
                 float*       __restrict__ row_sumsq) {
  const int wave = threadIdx.x >> 5;
  const int lane = threadIdx.x & 31;
  const int row  = blockIdx.x * (blockDim.x >> 5) + wave;   // 128 blocks * 8 waves = 1024

  const float4* __restrict__ xr =
      reinterpret_cast<const float4*>(x + (size_t)row * FEAT);
  const int lbl = labels[row];
  const float4* __restrict__ cr =
      reinterpret_cast<const float4*>(centers + (size_t)lbl * FEAT);

  float acc = 0.0f;
#pragma unroll
  for (int j = 0; j < FEAT / 4 / 32; ++j) {   // 4 iterations: 4x b128 per stream
    const float4 xv = xr[lane + 32 * j];
    const float4 cv = cr[lane + 32 * j];
    const float dx = xv.x - cv.x;
    const float dy = xv.y - cv.y;
    const float dz = xv.z - cv.z;
    const float dw = xv.w - cv.w;
    acc = fmaf(dx, dx, acc);
    acc = fmaf(dy, dy, acc);
    acc = fmaf(dz, dz, acc);
    acc = fmaf(dw, dw, acc);
  }

  const float tot = wave_reduce_sum_wmma(acc);   // EXEC all-ones: full wave, no divergence
  if (lane == 0) row_sumsq[row] = tot;
}

// Single wave folds the 1024 row partials to the scalar mean loss.
__global__ void __launch_bounds__(32)
center_loss_final(const float* __restrict__ row_sumsq,
                  float*       __restrict__ out) {
  const int lane = threadIdx.x & 31;
  const float4* __restrict__ p = reinterpret_cast<const float4*>(row_sumsq);

  float acc = 0.0f;
#pragma unroll
  for (int j = 0; j < BATCH / 4 / 32; ++j) {    // 8x b128 per lane
    const float4 v = p[lane + 32 * j];
    acc += (v.x + v.y) + (v.z + v.w);
  }

  const float tot = wave_reduce_sum_wmma(acc);
  if (lane == 0) out[0] = tot * (0.5f / (float)BATCH);
}

extern "C" void kernel_launch(void* const* d_in, const int* in_sizes, int n_in,
                              void* d_out, int out_size, void* d_ws, size_t ws_size,
                              hipStream_t stream) {
  (void)in_sizes; (void)n_in; (void)out_size; (void)ws_size;
  const float* x       = (const float*)d_in[0];
  const int*   labels  = (const int*)  d_in[1];
  const float* centers = (const float*)d_in[2];
  float*       ws      = (float*)d_ws;          // 1024 floats, fully rewritten each call
  float*       out     = (float*)d_out;

  center_loss_rows <<<BATCH / 8, 256, 0, stream>>>(x, labels, centers, ws);
  center_loss_final<<<1,         32,  0, stream>>>(ws, out);
}